// BiaffineScoreLayer_16604343566720
// MI455X (gfx1250) — compile-verified
//
#include <hip/hip_runtime.h>
#include <hip/hip_bf16.h>

// Problem constants (from reference): B=8, L=2048, DIM=512, CDIM=3
#define BATCH 8
#define LSEQ  2048
#define DIMC  512
#define CDIM  3
#define L1    (LSEQ - 1)              // 2047
#define ROWS  (BATCH * LSEQ)          // 16384
#define ROWS2 (BATCH * L1)            // 16376
#define UD    (DIMC + 1)              // 513
#define STR   516                     // padded LDS row stride (floats), 16B aligned, bank-spread

typedef __attribute__((ext_vector_type(2))) float v2f;
typedef __attribute__((ext_vector_type(8))) float v8f;

// -------------------------------------------------------------------------
// Kernel 1: fused gate GEMMs + sigmoid blend.
//   gatef = sigmoid((f+c)@Wgf + bgf);  fg = f*gatef + c*(1-gatef)
//   gateb = sigmoid((b+c)@Wgb + bgb);  bg = b*gateb + c*(1-gateb)
// One wave computes one 16x16 output tile of BOTH gemms via V_WMMA_F32_16X16X4_F32.
// grid = (1024 Mtiles, 8 Ngroups), block = 128 (4 waves, wave w -> N-subtile w)
// -------------------------------------------------------------------------
__global__ __launch_bounds__(128)
void gate_kernel(const float* __restrict__ f, const float* __restrict__ c,
                 const float* __restrict__ b,
                 const float* __restrict__ Wgf, const float* __restrict__ bgf,
                 const float* __restrict__ Wgb, const float* __restrict__ bgb,
                 float* __restrict__ fg, float* __restrict__ bg)
{
    const int lane  = threadIdx.x & 31;
    const int wave  = threadIdx.x >> 5;
    const int row0  = blockIdx.x * 16;
    const int ncol0 = (blockIdx.y * 4 + wave) * 16;

    // A fragment: lane holds A[m][k], m = lane&15, k = kbase + (lane>>4)*2 + {0,1}
    const int am = lane & 15;
    const int ak = (lane >> 4) << 1;
    const int nn = lane & 15;

    const float* frow = f + (size_t)(row0 + am) * DIMC;
    const float* crow = c + (size_t)(row0 + am) * DIMC;
    const float* brow = b + (size_t)(row0 + am) * DIMC;

    v8f accf = {}; v8f accb = {};

    for (int k = 0; k < DIMC; k += 4) {
        const int kk = k + ak;
        v2f fv = *(const v2f*)(frow + kk);
        v2f cv = *(const v2f*)(crow + kk);
        v2f bv = *(const v2f*)(brow + kk);
        v2f af = fv + cv;             // A for gatef gemm
        v2f ab = bv + cv;             // A for gateb gemm
        v2f wf, wb;                   // B fragment: B[k][n], n = lane&15
        wf.x = Wgf[(size_t)kk * DIMC + ncol0 + nn];
        wf.y = Wgf[(size_t)(kk + 1) * DIMC + ncol0 + nn];
        wb.x = Wgb[(size_t)kk * DIMC + ncol0 + nn];
        wb.y = Wgb[(size_t)(kk + 1) * DIMC + ncol0 + nn];
        accf = __builtin_amdgcn_wmma_f32_16x16x4_f32(false, af, false, wf,
                                                     (short)0, accf, false, false);
        accb = __builtin_amdgcn_wmma_f32_16x16x4_f32(false, ab, false, wb,
                                                     (short)0, accb, false, false);
    }

    // Epilogue: C/D layout => VGPR r holds (m = r + 8*(lane>>4), n = lane&15)
    const int gn    = ncol0 + (lane & 15);
    const float bf_ = bgf[gn];
    const float bb_ = bgb[gn];
    const int mh    = (lane >> 4) * 8;
#pragma unroll
    for (int r = 0; r < 8; ++r) {
        const int gm = row0 + mh + r;
        const size_t idx = (size_t)gm * DIMC + gn;
        float gfv = 1.0f / (1.0f + __expf(-(accf[r] + bf_)));
        float gbv = 1.0f / (1.0f + __expf(-(accb[r] + bb_)));
        float fv = f[idx], cv = c[idx], bv = b[idx];
        fg[idx] = fv * gfv + cv * (1.0f - gfv);
        bg[idx] = bv * gbv + cv * (1.0f - gbv);
    }
}

// -------------------------------------------------------------------------
// Kernel 2: biaffine scores.
//   asw[row,c] = f1 . U_c . b1, f1=[fg(row),1], b1=[bg(row+1),1]
// Split: bilinear core (512x512 of U_c) via WMMA with fg/bg tiles staged in LDS,
// plus border terms from the appended ones column. grid = (1024 Mtiles, 3 classes).
// -------------------------------------------------------------------------
__global__ __launch_bounds__(128)
void biaffine_kernel(const float* __restrict__ fg, const float* __restrict__ bg,
                     const float* __restrict__ U, float* __restrict__ asw)
{
    __shared__ float fgS[16 * STR];
    __shared__ float bgS[16 * STR];
    __shared__ float aswS[16];

    const int tid  = threadIdx.x;
    const int lane = tid & 31;
    const int wave = tid >> 5;
    const int row0 = blockIdx.x * 16;
    const int cc   = blockIdx.y;

    // Stage 16 x 512 tiles of fg (t) and bg (t+1) into LDS.
    {
        const int m   = tid >> 3;         // 16 rows, 8 threads each
        const int sub = tid & 7;          // 64 floats per thread
        const int row = row0 + m;
        const bool valid = row < ROWS2;
        const int bb = valid ? (row / L1) : 0;
        const int tt = valid ? (row - bb * L1) : 0;
        const float* fsrc = fg + (size_t)(bb * LSEQ + tt) * DIMC;
        const float* bsrc = bg + (size_t)(bb * LSEQ + tt + 1) * DIMC;
        for (int j = sub * 64; j < sub * 64 + 64; j += 4) {
            float4 fv = valid ? *(const float4*)(fsrc + j) : make_float4(0.f, 0.f, 0.f, 0.f);
            float4 bv = valid ? *(const float4*)(bsrc + j) : make_float4(0.f, 0.f, 0.f, 0.f);
            *(float4*)&fgS[m * STR + j] = fv;
            *(float4*)&bgS[m * STR + j] = bv;
        }
    }
    if (tid < 16) aswS[tid] = 0.0f;
    __syncthreads();

    const float* Uc = U + (size_t)cc * UD * UD;
    const int am = lane & 15;
    const int ak = (lane >> 4) << 1;
    const int nn = lane & 15;

    v8f aswAcc = {};

    for (int nt = wave; nt < 32; nt += 4) {           // 8 N-tiles per wave
        const int nbase = nt * 16;
        v8f acc = {};
        for (int k = 0; k < DIMC; k += 4) {
            const int kk = k + ak;
            v2f a = *(const v2f*)&fgS[am * STR + kk]; // A from LDS
            v2f w;                                    // B from U_c (L2-resident, 3.2MB)
            w.x = Uc[(size_t)kk * UD + nbase + nn];
            w.y = Uc[(size_t)(kk + 1) * UD + nbase + nn];
            acc = __builtin_amdgcn_wmma_f32_16x16x4_f32(false, a, false, w,
                                                        (short)0, acc, false, false);
        }
        const int mh = (lane >> 4) * 8;
#pragma unroll
        for (int r = 0; r < 8; ++r)                   // T[m][n] * bg[m][n], summed over n
            aswAcc[r] += acc[r] * bgS[(mh + r) * STR + nbase + nn];
    }

    // Reduce over the 16 n-lanes of each half-wave (xor masks stay inside the half).
#pragma unroll
    for (int r = 0; r < 8; ++r) {
        float v = aswAcc[r];
        v += __shfl_xor(v, 1);
        v += __shfl_xor(v, 2);
        v += __shfl_xor(v, 4);
        v += __shfl_xor(v, 8);
        if ((lane & 15) == 0) {
            const int m = (lane >> 4) * 8 + r;
            atomicAdd(&aswS[m], v);                   // ds_add_f32
        }
    }

    // Border terms from appended ones: fg . U[:,512] + U[512,:] . bg
    {
        const int m  = tid >> 3;
        const int i0 = (tid & 7) * 64;
        float t = 0.0f;
        for (int i = i0; i < i0 + 64; ++i)
            t += fgS[m * STR + i] * Uc[(size_t)i * UD + DIMC]
               + Uc[(size_t)DIMC * UD + i] * bgS[m * STR + i];
        atomicAdd(&aswS[m], t);
    }
    __syncthreads();

    if (tid < 16) {
        const int row = row0 + tid;
        if (row < ROWS2)
            asw[(size_t)row * CDIM + cc] = aswS[tid] + Uc[(size_t)DIMC * UD + DIMC];
    }
}

// -------------------------------------------------------------------------
// Kernel 3: score = argmax over the 3 classes (first-max tie-break, like jnp).
// -------------------------------------------------------------------------
__global__ __launch_bounds__(256)
void argmax_kernel(const float* __restrict__ asw, int* __restrict__ score)
{
    const int r = blockIdx.x * blockDim.x + threadIdx.x;
    if (r >= ROWS2) return;
    const float a0 = asw[(size_t)r * 3 + 0];
    const float a1 = asw[(size_t)r * 3 + 1];
    const float a2 = asw[(size_t)r * 3 + 2];
    int s = 0; float best = a0;
    if (a1 > best) { best = a1; s = 1; }
    if (a2 > best) { s = 2; }
    score[r] = s;
}

// -------------------------------------------------------------------------
// Kernel 4: per-batch segment cumsum. sc = [2, score...]; seg = cumsum(sc==2).
// Serial scan: 8 batches x 2048 steps, trivial.
// -------------------------------------------------------------------------
__global__ void seg_kernel(const int* __restrict__ score, int* __restrict__ seg)
{
    if (threadIdx.x != 0) return;
    const int b = blockIdx.x;
    int run = 0;
    for (int t = 0; t < LSEQ; ++t) {
        const int sc = (t == 0) ? 2 : score[b * L1 + (t - 1)];
        run += (sc == 2) ? 1 : 0;
        seg[b * LSEQ + t] = run;
    }
}

// -------------------------------------------------------------------------
// Kernel 5: masks[b,i,j] = mask[b,i] ? 0 : (seg[b,j] == seg[b,i])  (post-swapaxes)
// float4 stores, 134MB of traffic -> pure bandwidth.
// -------------------------------------------------------------------------
__global__ __launch_bounds__(256)
void masks_kernel(const int* __restrict__ seg, const unsigned char* __restrict__ mask,
                  float* __restrict__ out)
{
    const size_t gid = (size_t)blockIdx.x * blockDim.x + threadIdx.x;
    const int jp = (int)(gid % (LSEQ / 4));
    const size_t bi = gid / (LSEQ / 4);
    const int i = (int)(bi % LSEQ);
    const int b = (int)(bi / LSEQ);
    const int si = seg[b * LSEQ + i];
    const bool mi = mask[b * LSEQ + i] != 0;
    const int j = jp * 4;
    const int* sj = &seg[b * LSEQ + j];
    float4 r;
    r.x = (!mi && sj[0] == si) ? 1.0f : 0.0f;
    r.y = (!mi && sj[1] == si) ? 1.0f : 0.0f;
    r.z = (!mi && sj[2] == si) ? 1.0f : 0.0f;
    r.w = (!mi && sj[3] == si) ? 1.0f : 0.0f;
    *(float4*)&out[((size_t)b * LSEQ + i) * LSEQ + j] = r;
}

// -------------------------------------------------------------------------
// Kernel 6: al1/al2/al3. asw3 row = [asw1(t), asw2(t)] (6 values, shared per row),
// times tiny 6x512 weights. NOTE reference mapping: al1<-We1/be1, al2<-We3/be3,
// al3<-We2/be2; rows zeroed where mask[b,t].
// -------------------------------------------------------------------------
__global__ __launch_bounds__(256)
void al_kernel(const float* __restrict__ asw, const unsigned char* __restrict__ mask,
               const float* __restrict__ We1, const float* __restrict__ be1,
               const float* __restrict__ We2, const float* __restrict__ be2,
               const float* __restrict__ We3, const float* __restrict__ be3,
               float* __restrict__ al1, float* __restrict__ al2, float* __restrict__ al3)
{
    const int bt = blockIdx.x;            // 0..ROWS-1
    const int b  = bt >> 11;              // /2048
    const int t  = bt & (LSEQ - 1);
    __shared__ float row6[6];
    if (threadIdx.x < 3) {
        const int c = threadIdx.x;
        row6[c]     = (t < L1) ? asw[((size_t)(b * L1 + t)) * 3 + c] : 0.0f;
        row6[3 + c] = (t > 0)  ? asw[((size_t)(b * L1 + t - 1)) * 3 + c] : 0.0f;
    }
    __syncthreads();
    const bool mz = mask[b * LSEQ + t] != 0;
    for (int d = threadIdx.x; d < DIMC; d += 256) {
        float s1 = be1[d], s2 = be3[d], s3 = be2[d];
#pragma unroll
        for (int s = 0; s < 6; ++s) {
            const float a = row6[s];
            s1 += a * We1[s * DIMC + d];
            s2 += a * We3[s * DIMC + d];
            s3 += a * We2[s * DIMC + d];
        }
        const size_t o = (size_t)bt * DIMC + d;
        al1[o] = mz ? 0.0f : s1;
        al2[o] = mz ? 0.0f : s2;
        al3[o] = mz ? 0.0f : s3;
    }
}

// -------------------------------------------------------------------------
extern "C" void kernel_launch(void* const* d_in, const int* in_sizes, int n_in,
                              void* d_out, int out_size, void* d_ws, size_t ws_size,
                              hipStream_t stream)
{
    const float* f   = (const float*)d_in[0];
    const float* c   = (const float*)d_in[1];
    const float* b   = (const float*)d_in[2];
    const unsigned char* mask = (const unsigned char*)d_in[3];  // jnp bool -> 1 byte
    const float* Wgf = (const float*)d_in[4];
    const float* bgf = (const float*)d_in[5];
    const float* Wgb = (const float*)d_in[6];
    const float* bgb = (const float*)d_in[7];
    const float* U   = (const float*)d_in[8];
    const float* We1 = (const float*)d_in[9];
    const float* be1 = (const float*)d_in[10];
    const float* We2 = (const float*)d_in[11];
    const float* be2 = (const float*)d_in[12];
    const float* We3 = (const float*)d_in[13];
    const float* be3 = (const float*)d_in[14];

    // d_out layout: asw (ROWS2*3) | masks (B*L*L) | al1 | al2 | al3 (each ROWS*DIMC)
    float* out   = (float*)d_out;
    float* asw   = out;
    float* masks = out + (size_t)ROWS2 * CDIM;
    float* al1   = masks + (size_t)BATCH * LSEQ * LSEQ;
    float* al2   = al1 + (size_t)ROWS * DIMC;
    float* al3   = al2 + (size_t)ROWS * DIMC;

    // workspace: fg | bg | score | seg
    float* fg    = (float*)d_ws;
    float* bgw   = fg + (size_t)ROWS * DIMC;
    int*   score = (int*)(bgw + (size_t)ROWS * DIMC);
    int*   seg   = score + ROWS;   // score uses ROWS2 ints; ROWS for alignment slack

    gate_kernel<<<dim3(1024, 8), 128, 0, stream>>>(f, c, b, Wgf, bgf, Wgb, bgb, fg, bgw);
    biaffine_kernel<<<dim3(1024, CDIM), 128, 0, stream>>>(fg, bgw, U, asw);
    argmax_kernel<<<(ROWS2 + 255) / 256, 256, 0, stream>>>(asw, score);
    seg_kernel<<<BATCH, 32, 0, stream>>>(score, seg);
    masks_kernel<<<(BATCH * LSEQ * (LSEQ / 4)) / 256, 256, 0, stream>>>(seg, mask, masks);
    al_kernel<<<ROWS, 256, 0, stream>>>(asw, mask, We1, be1, We2, be2, We3, be3,
                                        al1, al2, al3);
}